// GarNet_7962869367454
// MI455X (gfx1250) — compile-verified
//
#include <hip/hip_runtime.h>
#include <math.h>

// GarNet, factored form for MI455X (gfx1250, wave32, WMMA).
//
// Naive reference materializes upd (B,V,6144) = 805 MB and runs a 52-GFLOP
// GEMM -> >=70us of HBM at 23.3 TB/s. We exploit upd[b,v,a*192+j] =
// ew[b,v,a]*agg[b,a,j] to collapse the 6192-wide GEMM into a per-batch
// (1024x48)@(48x128) GEMM => ~0.5 GFLOP, ~45 MB traffic, HBM-bound ~2-4us.
// All matrix math stays fp32 via V_WMMA_F32_16X16X4_F32 (exact vs reference).
//
// v2: removed all per-lane ternaries from the WMMA loops (they compiled to
// exec-mask branches between wmmas): weights are pre-packed into a combined
// WC matrix, and k4's x/ew K-split is done statically in the unrolled loop.

typedef __attribute__((ext_vector_type(2))) float v2f;
typedef __attribute__((ext_vector_type(8))) float v8f;

#define B_   32
#define V_   1024
#define F_   16
#define A_   32
#define P_   64
#define NF_  128
#define C_   96      // P + A, width of f = [feat | ew]
#define KTOT 48      // F + A, collapsed K of the output GEMM
#define ROWS (B_ * V_)

// WMMA f32 16x16x4 per-lane layout (ISA 7.12.2, 32-bit tables):
//   m/n = lane&15 ; kh = 2*(lane>>4)
//   A lane holds A[m][kh], A[m][kh+1] ; B lane holds B[kh][n], B[kh+1][n]
//   D vgpr r holds row (r + 8*(lane>>4)), col n.

// ---------------------------------------------------------------------------
// K0: pack WC (16 x 96) = [W_flr | W_s], bc (96) = [b_flr | b_s].
__global__ __launch_bounds__(256) void k0_prep(
    const float* __restrict__ Wf, const float* __restrict__ bf,
    const float* __restrict__ Ws, const float* __restrict__ bs,
    float* __restrict__ WC, float* __restrict__ bc)
{
    const int tid = blockIdx.x * 256 + threadIdx.x;
    if (tid < F_ * C_) {
        const int k = tid / C_, c = tid % C_;
        WC[tid] = (c < P_) ? Wf[k * P_ + c] : Ws[k * A_ + (c - P_)];
    } else if (tid < F_ * C_ + C_) {
        const int c = tid - F_ * C_;
        bc[c] = (c < P_) ? bf[c] : bs[c - P_];
    }
}

// ---------------------------------------------------------------------------
// K1: f[row, 0:64]  = x @ W_flr + b_flr
//     f[row,64:96]  = exp(-(x @ W_s + b_s)^2)
// One 16x16 tile per wave; 6 waves/block cover N=96; K=16 (4 WMMAs).
__global__ __launch_bounds__(192) void k1_vertex(
    const float* __restrict__ x,  const float* __restrict__ WC,
    const float* __restrict__ bc, float* __restrict__ fws)
{
    const int lane  = threadIdx.x & 31;
    const int ntile = threadIdx.x >> 5;          // 0..5
    const int mtile = blockIdx.x;                // 0..2047 (16 vertices each)
    const int col   = ntile * 16 + (lane & 15);  // 0..95
    const int kh    = (lane >> 4) * 2;
    const float* xr = x + (size_t)(mtile * 16 + (lane & 15)) * F_;

    v8f acc = {};
#pragma unroll
    for (int kt = 0; kt < 4; ++kt) {
        const int k0 = kt * 4 + kh;
        const v2f a = *(const v2f*)(xr + k0);          // global_load_b64
        v2f b;
        b.x = WC[k0 * C_ + col];
        b.y = WC[(k0 + 1) * C_ + col];
        acc = __builtin_amdgcn_wmma_f32_16x16x4_f32(
            false, a, false, b, (short)0, acc, false, false);
    }

    const float bias = bc[col];
#pragma unroll
    for (int r = 0; r < 8; ++r) {
        const int row = mtile * 16 + r + ((lane >> 4) * 8);
        float v = acc[r] + bias;
        if (col >= P_) v = expf(-v * v);          // edge weights (post-WMMA)
        fws[(size_t)row * C_ + col] = v;
    }
}

// ---------------------------------------------------------------------------
// K2: agg[b,a,0:96]  = max_v ew[b,v,a]*f[b,v,:]
//     agg[b,a,96:192]= mean_v ew[b,v,a]*f[b,v,:]
// One block per (b,a); thread = channel c (coalesced 96-wide rows of f).
__global__ __launch_bounds__(96) void k2_agg(
    const float* __restrict__ fws, float* __restrict__ agg)
{
    const int ba = blockIdx.x;                   // b*32 + a
    const int b  = ba >> 5, a = ba & 31;
    const int c  = threadIdx.x;                  // 0..95
    const float* fb = fws + (size_t)b * V_ * C_;

    float mx = -INFINITY, sm = 0.0f;
#pragma unroll 4
    for (int v = 0; v < V_; ++v) {
        const float w   = fb[(size_t)v * C_ + P_ + a];   // ew broadcast
        const float val = w * fb[(size_t)v * C_ + c];
        mx = fmaxf(mx, val);
        sm += val;
    }
    float* ao = agg + (size_t)ba * 192;
    ao[c]      = mx;
    ao[96 + c] = sm * (1.0f / (float)V_);
}

// ---------------------------------------------------------------------------
// K3: per-batch collapsed B matrix BW[b] (48 x 128):
//     rows 0..15  = W_out[0:16,:]                       (x block)
//     rows 16..47 = agg[b,a,:] @ W_out[16+a*192:+192,:] + W_out[6160+a,:]
__global__ __launch_bounds__(128) void k3_bw(
    const float* __restrict__ agg, const float* __restrict__ Wout,
    float* __restrict__ BW)
{
    const int bid  = blockIdx.x;                 // b*48 + krow
    const int b    = bid / KTOT, krow = bid % KTOT;
    const int n    = threadIdx.x;                // 0..127
    float acc;
    if (krow < F_) {
        acc = Wout[(size_t)krow * NF_ + n];
    } else {
        const int a = krow - F_;
        acc = Wout[(size_t)(F_ + 2 * A_ * C_ + a) * NF_ + n]; // ew row 6160+a
        const float* ag = agg + (size_t)(b * A_ + a) * 192;
        const float* wr = Wout + (size_t)(F_ + a * 192) * NF_ + n;
#pragma unroll 4
        for (int c = 0; c < 192; ++c)
            acc = fmaf(ag[c], wr[(size_t)c * NF_], acc);
    }
    BW[(size_t)bid * NF_ + n] = acc;
}

// ---------------------------------------------------------------------------
// K4: out[b] = tanh([x[b] | ew[b]] (1024x48) @ BW[b] (48x128) + b_out)
// Block = 16 vertex rows x full N=128 (8 waves, one 16x16 tile each); K=48.
// The x/ew split at K=16 is 4-aligned => handled statically, no branches.
__global__ __launch_bounds__(256) void k4_out(
    const float* __restrict__ x,   const float* __restrict__ fws,
    const float* __restrict__ BW,  const float* __restrict__ bout,
    float* __restrict__ out)
{
    const int lane  = threadIdx.x & 31;
    const int ntile = threadIdx.x >> 5;          // 0..7
    const int bid   = blockIdx.x;                // b*64 + mtile
    const int b     = bid >> 6, mtile = bid & 63;
    const int col   = ntile * 16 + (lane & 15);
    const int kh    = (lane >> 4) * 2;
    const int rowA  = b * V_ + mtile * 16 + (lane & 15);

    const float* xr  = x   + (size_t)rowA * F_;
    const float* er  = fws + (size_t)rowA * C_ + P_;   // ew columns of f
    const float* bwb = BW  + (size_t)b * KTOT * NF_;

    v8f acc = {};
#pragma unroll
    for (int kt = 0; kt < 4; ++kt) {             // K 0..15 : x block
        const int k0 = kt * 4 + kh;
        const v2f a = *(const v2f*)(xr + k0);
        v2f bb;
        bb.x = bwb[(size_t)k0 * NF_ + col];
        bb.y = bwb[(size_t)(k0 + 1) * NF_ + col];
        acc = __builtin_amdgcn_wmma_f32_16x16x4_f32(
            false, a, false, bb, (short)0, acc, false, false);
    }
#pragma unroll
    for (int kt = 4; kt < 12; ++kt) {            // K 16..47 : ew block
        const int k0 = kt * 4 + kh;
        const v2f a = *(const v2f*)(er + (k0 - F_));
        v2f bb;
        bb.x = bwb[(size_t)k0 * NF_ + col];
        bb.y = bwb[(size_t)(k0 + 1) * NF_ + col];
        acc = __builtin_amdgcn_wmma_f32_16x16x4_f32(
            false, a, false, bb, (short)0, acc, false, false);
    }

    const float bias = bout[col];
#pragma unroll
    for (int r = 0; r < 8; ++r) {
        const int row = b * V_ + mtile * 16 + r + ((lane >> 4) * 8);
        out[(size_t)row * NF_ + col] = tanhf(acc[r] + bias);
    }
}

// ---------------------------------------------------------------------------
extern "C" void kernel_launch(void* const* d_in, const int* in_sizes, int n_in,
                              void* d_out, int out_size, void* d_ws, size_t ws_size,
                              hipStream_t stream)
{
    const float* x    = (const float*)d_in[0];
    const float* Wf   = (const float*)d_in[1];
    const float* bf   = (const float*)d_in[2];
    const float* Ws   = (const float*)d_in[3];
    const float* bs   = (const float*)d_in[4];
    const float* Wout = (const float*)d_in[5];
    const float* bout = (const float*)d_in[6];
    float* out = (float*)d_out;

    // Workspace layout (total ~14.2 MB):
    //   fws : ROWS*96  floats (12.58 MB)   f = [feat | ew]
    //   agg : B*A*192  floats ( 0.79 MB)   [max | mean]
    //   BW  : B*48*128 floats ( 0.79 MB)   collapsed per-batch B matrix
    //   WC  : 16*96    floats ( 6 KB)      packed [W_flr | W_s]
    //   bc  : 96       floats              packed [b_flr | b_s]
    float* fws = (float*)d_ws;
    float* agg = fws + (size_t)ROWS * C_;
    float* BW  = agg + (size_t)B_ * A_ * 192;
    float* WC  = BW  + (size_t)B_ * KTOT * NF_;
    float* bc  = WC  + (size_t)F_ * C_;

    k0_prep  <<<7,         256, 0, stream>>>(Wf, bf, Ws, bs, WC, bc);
    k1_vertex<<<ROWS / 16, 192, 0, stream>>>(x, WC, bc, fws);
    k2_agg   <<<B_ * A_,   96,  0, stream>>>(fws, agg);
    k3_bw    <<<B_ * KTOT, 128, 0, stream>>>(agg, Wout, BW);
    k4_out   <<<B_ * 64,   256, 0, stream>>>(x, fws, BW, bout, out);
}